// GATLayer_23673859736033
// MI455X (gfx1250) — compile-verified
//
#include <hip/hip_runtime.h>
#include <hip/hip_bf16.h>
#include <math.h>

#define IN_DIM   128
#define HEADS    4
#define OUT_DIM  32
#define HC       128   // HEADS*OUT_DIM
#define NEG_SLOPE 0.04821699482733621f
#define BN_EPS    1e-5f

typedef float v2f __attribute__((ext_vector_type(2)));
typedef float v8f __attribute__((ext_vector_type(8)));

// ---------------------------------------------------------------------------
// float atomic max via integer punning (safe, no CAS loop):
//   non-negative floats are monotone under signed-int compare,
//   negative floats are monotone (reversed) under unsigned compare.
__device__ __forceinline__ void atomicMaxF(float* a, float v) {
    if (!(v < 0.0f)) {
        atomicMax((int*)a, __float_as_int(v));
    } else {
        atomicMin((unsigned int*)a, (unsigned int)__float_as_int(v));
    }
}

// ---------------------------------------------------------------------------
// 0) init: zero out, colsum/colsumsq, denom; set running-max to -inf
__global__ void init_kernel(float* __restrict__ out, float* __restrict__ mmax,
                            float* __restrict__ denom, float* __restrict__ colsum,
                            float* __restrict__ colsumsq, int N) {
    long long tid    = blockIdx.x * (long long)blockDim.x + threadIdx.x;
    long long stride = (long long)gridDim.x * blockDim.x;
    long long totHC  = (long long)N * HC;
    for (long long i = tid; i < totHC; i += stride) out[i] = 0.0f;
    long long totH = (long long)N * HEADS;
    for (long long i = tid; i < totH; i += stride) {
        mmax[i]  = -INFINITY;
        denom[i] = 0.0f;
    }
    if (tid < HC) { colsum[tid] = 0.0f; colsumsq[tid] = 0.0f; }
}

// ---------------------------------------------------------------------------
// 1) h = x @ W  via V_WMMA_F32_16X16X4_F32.
//    Block = 256 threads = 8 waves. Wave w computes the 16x16 tile at
//    (rows blockIdx.x*16 .. +15, cols w*16 .. +15). K = 128 in steps of 4.
__global__ void __launch_bounds__(256)
gemm_h_wmma(const float* __restrict__ x, const float* __restrict__ W,
            float* __restrict__ h, int N) {
    const int wave = threadIdx.x >> 5;          // 0..7 -> column tile
    const int lane = threadIdx.x & 31;
    const int row0 = blockIdx.x * 16;           // N is a multiple of 16 (50000)
    const int m    = lane & 15;                 // A row within tile
    const int kb   = (lane >> 4) << 1;          // 0 (lanes 0-15) or 2 (lanes 16-31)
    const int n    = (wave << 4) + (lane & 15); // output column

    v8f acc = {};
    const float* xr = x + (size_t)(row0 + m) * IN_DIM;
    #pragma unroll 4
    for (int k = 0; k < IN_DIM; k += 4) {
        v2f a, b;
        a.x = xr[k + kb];
        a.y = xr[k + kb + 1];
        b.x = W[(size_t)(k + kb)     * HC + n];
        b.y = W[(size_t)(k + kb + 1) * HC + n];
        acc = __builtin_amdgcn_wmma_f32_16x16x4_f32(
                  /*neg_a=*/false, a, /*neg_b=*/false, b,
                  /*c_mod=*/(short)0, acc, /*reuse_a=*/false, /*reuse_b=*/false);
    }
    // D layout: VGPR r holds row (r + 8*(lane/16)), col = n
    const int mr = (lane >> 4) << 3;
    #pragma unroll
    for (int r = 0; r < 8; ++r) {
        h[(size_t)(row0 + mr + r) * HC + n] = acc[r];
    }
}

// ---------------------------------------------------------------------------
// 2) per-node attention logits: alpha_s[n,hd], alpha_d[n,hd]
__global__ void alpha_kernel(const float* __restrict__ h,
                             const float* __restrict__ a_src,
                             const float* __restrict__ a_dst,
                             float* __restrict__ as_, float* __restrict__ ad_,
                             int N) {
    int t = blockIdx.x * blockDim.x + threadIdx.x;
    if (t >= N * HEADS) return;
    const int n  = t >> 2;
    const int hd = t & 3;
    const float* hp  = h     + (size_t)n * HC + hd * OUT_DIM;
    const float* sp  = a_src + hd * OUT_DIM;
    const float* dp  = a_dst + hd * OUT_DIM;
    float s = 0.0f, d = 0.0f;
    #pragma unroll
    for (int c = 0; c < OUT_DIM; ++c) {
        float v = hp[c];
        s = fmaf(v, sp[c], s);
        d = fmaf(v, dp[c], d);
    }
    as_[t] = s;
    ad_[t] = d;
}

// ---------------------------------------------------------------------------
// 3) edge pass 1: raw scores (LeakyReLU) + segment max (per dst,head)
__global__ void edge_score_max(const long long* __restrict__ ei,
                               const float* __restrict__ as_,
                               const float* __restrict__ ad_,
                               float* __restrict__ ework, float* __restrict__ mmax,
                               long long E, int N) {
    long long i = blockIdx.x * (long long)blockDim.x + threadIdx.x;
    long long T = E + N;
    if (i >= T) return;
    long long s = (i < E) ? ei[i]     : (i - E);   // self-loop tail
    long long d = (i < E) ? ei[E + i] : (i - E);
    #pragma unroll
    for (int hd = 0; hd < HEADS; ++hd) {
        float e = as_[s * HEADS + hd] + ad_[d * HEADS + hd];
        e = (e >= 0.0f) ? e : NEG_SLOPE * e;
        ework[i * HEADS + hd] = e;
        atomicMaxF(&mmax[d * HEADS + hd], e);
    }
}

// 4) edge pass 2: exp(e - max) + segment sum (denominator)
__global__ void edge_exp_sum(const long long* __restrict__ ei,
                             const float* __restrict__ mmax,
                             float* __restrict__ ework, float* __restrict__ denom,
                             long long E, int N) {
    long long i = blockIdx.x * (long long)blockDim.x + threadIdx.x;
    long long T = E + N;
    if (i >= T) return;
    long long d = (i < E) ? ei[E + i] : (i - E);
    #pragma unroll
    for (int hd = 0; hd < HEADS; ++hd) {
        float w = __expf(ework[i * HEADS + hd] - mmax[d * HEADS + hd]);
        ework[i * HEADS + hd] = w;
        atomicAdd(&denom[d * HEADS + hd], w);
    }
}

// 5) edge pass 3: normalized weighted scatter-add of messages.
//    One thread per (edge, channel): coalesced h[src] reads, one atomic each.
__global__ void edge_aggregate(const long long* __restrict__ ei,
                               const float* __restrict__ h,
                               const float* __restrict__ ework,
                               const float* __restrict__ denom,
                               float* __restrict__ out,
                               long long E, int N) {
    long long t = blockIdx.x * (long long)blockDim.x + threadIdx.x;
    long long T = (E + N) * (long long)HC;
    if (t >= T) return;
    long long i  = t >> 7;            // edge id
    int       c  = (int)(t & 127);    // channel 0..127
    int       hd = c >> 5;            // head
    long long s = (i < E) ? ei[i]     : (i - E);
    long long d = (i < E) ? ei[E + i] : (i - E);
    float alpha = ework[i * HEADS + hd] / denom[d * HEADS + hd];
    atomicAdd(&out[d * HC + c], h[s * HC + c] * alpha);
}

// ---------------------------------------------------------------------------
// 6) bias add + per-column sum / sumsq (block-local accumulation, few atomics)
__global__ void __launch_bounds__(128)
bn_reduce(float* __restrict__ out, const float* __restrict__ bias,
          float* __restrict__ colsum, float* __restrict__ colsumsq, int N) {
    const int c = threadIdx.x;        // 128 threads = 128 columns
    const float b = bias[c];
    float s = 0.0f, sq = 0.0f;
    for (int n = blockIdx.x; n < N; n += gridDim.x) {
        float v = out[(size_t)n * HC + c] + b;
        out[(size_t)n * HC + c] = v;
        s += v;
        sq = fmaf(v, v, sq);
    }
    atomicAdd(&colsum[c], s);
    atomicAdd(&colsumsq[c], sq);
}

// 7) batch-norm normalize
__global__ void bn_normalize(float* __restrict__ out,
                             const float* __restrict__ colsum,
                             const float* __restrict__ colsumsq,
                             const float* __restrict__ gamma,
                             const float* __restrict__ beta, int N) {
    long long t = blockIdx.x * (long long)blockDim.x + threadIdx.x;
    if (t >= (long long)N * HC) return;
    int c = (int)(t & 127);
    float invN = 1.0f / (float)N;
    float mu   = colsum[c] * invN;
    float var  = colsumsq[c] * invN - mu * mu;
    out[t] = (out[t] - mu) * rsqrtf(var + BN_EPS) * gamma[c] + beta[c];
}

// ---------------------------------------------------------------------------
extern "C" void kernel_launch(void* const* d_in, const int* in_sizes, int n_in,
                              void* d_out, int out_size, void* d_ws, size_t ws_size,
                              hipStream_t stream) {
    const float*     x      = (const float*)d_in[0];
    const long long* ei     = (const long long*)d_in[1];
    const float*     W      = (const float*)d_in[2];
    const float*     a_src  = (const float*)d_in[3];
    const float*     a_dst  = (const float*)d_in[4];
    const float*     bias   = (const float*)d_in[5];
    const float*     gamma  = (const float*)d_in[6];
    const float*     beta   = (const float*)d_in[7];
    float*           out    = (float*)d_out;

    const int       N = in_sizes[0] / IN_DIM;          // 50000
    const long long E = (long long)in_sizes[1] / 2;    // 1600000
    const long long T = E + N;                         // edges + self loops

    // workspace layout (floats)
    float* ws      = (float*)d_ws;
    float* h       = ws;                                // N*128
    float* as_     = h     + (size_t)N * HC;            // N*4
    float* ad_     = as_   + (size_t)N * HEADS;         // N*4
    float* mmax    = ad_   + (size_t)N * HEADS;         // N*4
    float* denom   = mmax  + (size_t)N * HEADS;         // N*4
    float* ework   = denom + (size_t)N * HEADS;         // (E+N)*4
    float* colsum  = ework + (size_t)T * HEADS;         // 128
    float* colsumq = colsum + HC;                       // 128

    // 0) init accumulators
    init_kernel<<<2048, 256, 0, stream>>>(out, mmax, denom, colsum, colsumq, N);

    // 1) h = x @ W  (WMMA f32 16x16x4); N is an exact multiple of 16
    gemm_h_wmma<<<N / 16, 256, 0, stream>>>(x, W, h, N);

    // 2) attention logits
    alpha_kernel<<<(N * HEADS + 255) / 256, 256, 0, stream>>>(h, a_src, a_dst, as_, ad_, N);

    // 3-5) segment softmax + aggregation
    edge_score_max<<<(unsigned)((T + 255) / 256), 256, 0, stream>>>(ei, as_, ad_, ework, mmax, E, N);
    edge_exp_sum  <<<(unsigned)((T + 255) / 256), 256, 0, stream>>>(ei, mmax, ework, denom, E, N);
    edge_aggregate<<<(unsigned)((T * HC + 255) / 256), 256, 0, stream>>>(ei, h, ework, denom, out, E, N);

    // 6-7) bias + batch norm
    bn_reduce<<<512, 128, 0, stream>>>(out, bias, colsum, colsumq, N);
    bn_normalize<<<(unsigned)(((long long)N * HC + 255) / 256), 256, 0, stream>>>(
        out, colsum, colsumq, gamma, beta, N);
}